// MultiScaleNeighborConv_27273042330215
// MI455X (gfx1250) — compile-verified
//
#include <hip/hip_runtime.h>
#include <hip/hip_bf16.h>

#define B_  16
#define F1_ 32
#define C_  62
#define T_  1024
#define NT  256          // t-tile per workgroup
#define SA_STRIDE 36     // padded k-stride (words) -> conflict-free b64 reads
#define SX_STRIDE 36

typedef __attribute__((ext_vector_type(2))) float v2f;
typedef __attribute__((ext_vector_type(8))) float v8f;

__global__ __launch_bounds__(256)
void msnc_wmma_kernel(const float* __restrict__ x,
                      const int*   __restrict__ nbr3,
                      const int*   __restrict__ nbr5,
                      const float* __restrict__ W3,
                      const float* __restrict__ b3,
                      const float* __restrict__ W5,
                      const float* __restrict__ b5,
                      const float* __restrict__ alog,
                      float* __restrict__ y)
{
    __shared__ float sA[F1_ * SA_STRIDE];   // [g][k=f]   scaled weight tile (current neighbor)
    __shared__ float sX[NT * SX_STRIDE];    // [n][k=f]   gathered x tile, transposed
    __shared__ float sBias[F1_];

    const int tid    = threadIdx.x;
    const int lane   = tid & 31;
    const int wv     = tid >> 5;
    const int lane16 = lane & 15;
    const int hi     = lane >> 4;       // 0: lanes 0-15, 1: lanes 16-31
    const int koff   = hi * 2;          // per ISA A/B fragment layout (K split across half-waves)

    const int t0 = blockIdx.x * NT;
    const int b  = blockIdx.y;
    const int c  = blockIdx.z;

    // per-channel softmax over the two logits (uniform; redundant per-thread)
    const float l0 = alog[c];
    const float l1 = alog[C_ + c];
    const float mx = fmaxf(l0, l1);
    const float e0 = __expf(l0 - mx);
    const float e1 = __expf(l1 - mx);
    const float inv = 1.0f / (e0 + e1);
    const float a0s = e0 * inv;
    const float a1s = e1 * inv;

    if (tid < F1_) sBias[tid] = a0s * b3[tid] + a1s * b5[tid];
    __syncthreads();

    // accumulators: [g-half][n-subtile], each a 16x16 f32 C/D tile (8 VGPRs)
    // init with blended bias per output row g (C/D layout: VGPR j -> M = j + 8*hi)
    v8f acc[2][2];
    #pragma unroll
    for (int mi = 0; mi < 2; ++mi) {
        #pragma unroll
        for (int j = 0; j < 8; ++j) {
            const float bv = sBias[mi * 16 + hi * 8 + j];
            acc[mi][0][j] = bv;
            acc[mi][1][j] = bv;
        }
    }

    const int tt = t0 + tid;   // this thread's global t column (n = tid)

    #pragma unroll 1
    for (int m = 0; m < 8; ++m) {
        const int cm = (m < 3) ? nbr3[c * 3 + m] : nbr5[c * 5 + (m - 3)];
        __syncthreads();   // protect sA/sX from previous iteration's readers

        // ---- build alpha-scaled weight tile sA[g][f] (A matrix, K-slice of this neighbor) ----
        {
            const int g  = tid >> 3;         // 0..31
            const int f0 = (tid & 7) * 4;    // 0,4,...,28
            float4 w;
            if (m < 3) {
                const float* p = W3 + (g * F1_ + f0) * 3 + m;
                w.x = a0s * p[0]; w.y = a0s * p[3]; w.z = a0s * p[6]; w.w = a0s * p[9];
            } else {
                const float* p = W5 + (g * F1_ + f0) * 5 + (m - 3);
                w.x = a1s * p[0]; w.y = a1s * p[5]; w.z = a1s * p[10]; w.w = a1s * p[15];
            }
            *(float4*)&sA[g * SA_STRIDE + f0] = w;
        }

        // ---- gather x slab for neighbor channel cm, store transposed sX[n][f] ----
        {
            const float* xp = x + ((size_t)(b * F1_) * C_ + cm) * T_ + tt;
            float v[F1_];
            #pragma unroll
            for (int f = 0; f < F1_; ++f)
                v[f] = xp[(size_t)f * (C_ * T_)];    // coalesced: lane-consecutive t
            #pragma unroll
            for (int j = 0; j < F1_ / 4; ++j) {
                float4 q = make_float4(v[4*j], v[4*j+1], v[4*j+2], v[4*j+3]);
                *(float4*)&sX[tid * SX_STRIDE + 4 * j] = q;   // aligned b128 LDS store
            }
        }
        __syncthreads();

        // ---- 32 x 256 GEMM slice, K = 32, via v_wmma_f32_16x16x4_f32 ----
        #pragma unroll
        for (int s = 0; s < 8; ++s) {
            const int k0 = s * 4 + koff;
            v2f af[2], bf[2];
            af[0] = *(const v2f*)&sA[ lane16        * SA_STRIDE + k0];
            af[1] = *(const v2f*)&sA[(16 + lane16)  * SA_STRIDE + k0];
            bf[0] = *(const v2f*)&sX[(wv * 32 +      lane16) * SX_STRIDE + k0];
            bf[1] = *(const v2f*)&sX[(wv * 32 + 16 + lane16) * SX_STRIDE + k0];
            #pragma unroll
            for (int mi = 0; mi < 2; ++mi)
                #pragma unroll
                for (int ni = 0; ni < 2; ++ni)
                    acc[mi][ni] = __builtin_amdgcn_wmma_f32_16x16x4_f32(
                        false, af[mi], false, bf[ni],
                        (short)0, acc[mi][ni], false, false);
        }
    }

    // ---- epilogue: write y[b, g, c, t] ----
    #pragma unroll
    for (int mi = 0; mi < 2; ++mi) {
        #pragma unroll
        for (int ni = 0; ni < 2; ++ni) {
            const int tcol = t0 + wv * 32 + ni * 16 + lane16;
            #pragma unroll
            for (int j = 0; j < 8; ++j) {
                const int g = mi * 16 + hi * 8 + j;
                y[(((size_t)b * F1_ + g) * C_ + c) * T_ + tcol] = acc[mi][ni][j];
            }
        }
    }
}

extern "C" void kernel_launch(void* const* d_in, const int* in_sizes, int n_in,
                              void* d_out, int out_size, void* d_ws, size_t ws_size,
                              hipStream_t stream) {
    const float* x    = (const float*)d_in[0];
    const int*   nbr3 = (const int*)  d_in[1];
    const int*   nbr5 = (const int*)  d_in[2];
    const float* W3   = (const float*)d_in[3];
    const float* b3   = (const float*)d_in[4];
    const float* W5   = (const float*)d_in[5];
    const float* b5   = (const float*)d_in[6];
    const float* al   = (const float*)d_in[7];
    float* y = (float*)d_out;

    dim3 grid(T_ / NT, B_, C_);   // (4, 16, 62)
    msnc_wmma_kernel<<<grid, 256, 0, stream>>>(x, nbr3, nbr5, W3, b3, W5, b5, al, y);
}